// TransitionDown_56916906606935
// MI455X (gfx1250) — compile-verified
//
#include <hip/hip_runtime.h>

#define N_PTS   32768
#define C_IN    64
#define C_OUT   128
#define KNN_K   16
#define M_PTS   (N_PTS / 4)     // 8192
#define LN_EPS  1e-5f

typedef float v2f __attribute__((ext_vector_type(2)));
typedef float v4f __attribute__((ext_vector_type(4)));
typedef float v8f __attribute__((ext_vector_type(8)));

// ---------------------------------------------------------------------------
// Kernel 1: farthest point sampling. Single 1024-thread workgroup, sequential
// over M-1 steps. dmin held in registers (32 pts/thread), pos streamed from L2.
// Writes sub_pos[M*3] and sub_batch[M] directly (ids never materialized).
// ---------------------------------------------------------------------------
__global__ __launch_bounds__(1024, 1)
void fps_kernel(const float* __restrict__ pos,
                float* __restrict__ sub_pos,
                int* __restrict__ sub_batch)
{
    const int t = threadIdx.x;          // 0..1023
    const int wid = t >> 5, lane = t & 31;

    __shared__ float swv[32];
    __shared__ int   swi[32];
    __shared__ int   s_next;

    // sub_batch is all zeros (single cloud)
    for (int i = t; i < M_PTS; i += 1024) sub_batch[i] = 0;
    if (t == 0) {
        sub_pos[0] = pos[0];
        sub_pos[1] = pos[1];
        sub_pos[2] = pos[2];
    }

    float dmin[32];
#pragma unroll
    for (int j = 0; j < 32; ++j) dmin[j] = __builtin_inff();

    int last = 0;
    for (int step = 1; step < M_PTS; ++step) {
        const float lx = pos[3 * last + 0];
        const float ly = pos[3 * last + 1];
        const float lz = pos[3 * last + 2];

        float bv = -1.0f;
        int bidx = 0;
#pragma unroll
        for (int j = 0; j < 32; ++j) {
            const int i = t + j * 1024;
            const float dx = pos[3 * i + 0] - lx;
            const float dy = pos[3 * i + 1] - ly;
            const float dz = pos[3 * i + 2] - lz;
            const float d = dx * dx + dy * dy + dz * dz;
            dmin[j] = fminf(dmin[j], d);
            if (dmin[j] > bv) { bv = dmin[j]; bidx = i; }
        }
        // wave32 argmax reduce (ties -> lower index, matching jnp.argmax)
        for (int off = 16; off; off >>= 1) {
            const float ov = __shfl_down(bv, off, 32);
            const int   oi = __shfl_down(bidx, off, 32);
            if (ov > bv || (ov == bv && oi < bidx)) { bv = ov; bidx = oi; }
        }
        if (lane == 0) { swv[wid] = bv; swi[wid] = bidx; }
        __syncthreads();
        if (wid == 0) {
            bv = swv[lane]; bidx = swi[lane];
            for (int off = 16; off; off >>= 1) {
                const float ov = __shfl_down(bv, off, 32);
                const int   oi = __shfl_down(bidx, off, 32);
                if (ov > bv || (ov == bv && oi < bidx)) { bv = ov; bidx = oi; }
            }
            if (lane == 0) {
                s_next = bidx;
                sub_pos[3 * step + 0] = pos[3 * bidx + 0];
                sub_pos[3 * step + 1] = pos[3 * bidx + 1];
                sub_pos[3 * step + 2] = pos[3 * bidx + 2];
            }
        }
        __syncthreads();
        last = s_next;
    }
}

// ---------------------------------------------------------------------------
// Kernel 2: h = relu(LN(x @ W + b)) via V_WMMA_F32_16X16X4_F32.
// Block = 256 threads = 8 waves. Each wave: 16 rows x 128 cols (8 C tiles,
// K-loop of 16 x4 steps). W staged in LDS (32 KB). LayerNorm reduced inside
// the wave via __shfl_xor over the 16-lane halves of the C/D layout.
// ---------------------------------------------------------------------------
__global__ __launch_bounds__(256)
void mlp_ln_relu_kernel(const float* __restrict__ x,
                        const float* __restrict__ W,
                        const float* __restrict__ bias,
                        const float* __restrict__ gamma,
                        const float* __restrict__ beta,
                        float* __restrict__ h)
{
    __shared__ float sW[C_IN * C_OUT];   // 32 KB
    const int tid = threadIdx.x;
    for (int i = tid; i < (C_IN * C_OUT) / 4; i += 256)
        ((v4f*)sW)[i] = ((const v4f*)W)[i];
    __syncthreads();

    const int wave = tid >> 5;
    const int lane = tid & 31;
    const int sub  = lane & 15;
    const int half = lane >> 4;          // 0: K={0,1}, rows M=0..7 ; 1: K={2,3}, rows M=8..15
    const int r0   = (blockIdx.x * 8 + wave) * 16;
    const int koff = half * 2;

    v8f acc[8] = {};

    for (int k = 0; k < 16; ++k) {
        // A (16x4 f32): lane half selects K pair; contiguous float2 load
        const v2f a = *(const v2f*)(x + (r0 + sub) * C_IN + 4 * k + koff);
#pragma unroll
        for (int t = 0; t < 8; ++t) {
            const int col = t * 16 + sub;
            v2f bb;
            bb.x = sW[(4 * k + koff + 0) * C_OUT + col];
            bb.y = sW[(4 * k + koff + 1) * C_OUT + col];
            acc[t] = __builtin_amdgcn_wmma_f32_16x16x4_f32(
                false, a, false, bb, (short)0, acc[t], false, false);
        }
    }

    // + bias (per output channel)
#pragma unroll
    for (int t = 0; t < 8; ++t) {
        const float bv = bias[t * 16 + sub];
#pragma unroll
        for (int j = 0; j < 8; ++j) acc[t][j] += bv;
    }

    // LayerNorm over the 128 channels of each row (row j lives in C-VGPR j
    // across one 16-lane half). Reduce channel partials with xor-shuffles.
    float sum[8], ssq[8];
#pragma unroll
    for (int j = 0; j < 8; ++j) { sum[j] = 0.0f; ssq[j] = 0.0f; }
#pragma unroll
    for (int t = 0; t < 8; ++t)
#pragma unroll
        for (int j = 0; j < 8; ++j) {
            const float v = acc[t][j];
            sum[j] += v; ssq[j] += v * v;
        }
#pragma unroll
    for (int off = 1; off < 16; off <<= 1)
#pragma unroll
        for (int j = 0; j < 8; ++j) {
            sum[j] += __shfl_xor(sum[j], off, 32);
            ssq[j] += __shfl_xor(ssq[j], off, 32);
        }

    float mu[8], rstd[8];
#pragma unroll
    for (int j = 0; j < 8; ++j) {
        mu[j] = sum[j] * (1.0f / C_OUT);
        const float var = ssq[j] * (1.0f / C_OUT) - mu[j] * mu[j];
        rstd[j] = rsqrtf(var + LN_EPS);
    }

#pragma unroll
    for (int t = 0; t < 8; ++t) {
        const int col = t * 16 + sub;
        const float g = gamma[col];
        const float be = beta[col];
#pragma unroll
        for (int j = 0; j < 8; ++j) {
            const int row = r0 + j + half * 8;
            const float v = (acc[t][j] - mu[j]) * rstd[j] * g + be;
            h[row * C_OUT + col] = fmaxf(v, 0.0f);
        }
    }
}

// ---------------------------------------------------------------------------
// Kernel 3: fused KNN (top-16 over all N points) + gather-max pooling.
// One wave per query. Per-lane sorted top-16 over 1024 strided points, then
// 16-round cooperative min-extract over 512 LDS candidates, then max over the
// 16 neighbors' 128 channels (4 channels/lane, 16B loads).
// ---------------------------------------------------------------------------
__global__ __launch_bounds__(256)
void knn_pool_kernel(const float* __restrict__ pos,
                     const float* __restrict__ sub_pos,
                     const float* __restrict__ h,
                     float* __restrict__ out)
{
    __shared__ float sd[8][32 * KNN_K];
    __shared__ int   si[8][32 * KNN_K];
    __shared__ int   snbr[8][KNN_K];

    const int w = threadIdx.x >> 5;
    const int lane = threadIdx.x & 31;
    const int q = blockIdx.x * 8 + w;

    const float qx = sub_pos[3 * q + 0];
    const float qy = sub_pos[3 * q + 1];
    const float qz = sub_pos[3 * q + 2];

    float best[KNN_K];
    int besti[KNN_K];
#pragma unroll
    for (int j = 0; j < KNN_K; ++j) { best[j] = __builtin_inff(); besti[j] = -1; }

    for (int i = lane; i < N_PTS; i += 32) {
        __builtin_prefetch(pos + 3 * (i + 256), 0, 0);
        const float dx = pos[3 * i + 0] - qx;
        const float dy = pos[3 * i + 1] - qy;
        const float dz = pos[3 * i + 2] - qz;
        const float d2 = dx * dx + dy * dy + dz * dz;
        if (d2 < best[KNN_K - 1]) {
            float v = d2; int vi = i;
#pragma unroll
            for (int j = 0; j < KNN_K; ++j) {
                if (v < best[j]) {
                    const float tv = best[j]; const int ti = besti[j];
                    best[j] = v; besti[j] = vi;
                    v = tv; vi = ti;
                }
            }
        }
    }

#pragma unroll
    for (int j = 0; j < KNN_K; ++j) {
        sd[w][lane + 32 * j] = best[j];
        si[w][lane + 32 * j] = besti[j];
    }
    __syncthreads();

    // 16 rounds of cooperative min-extract over the 512 candidates
    for (int r = 0; r < KNN_K; ++r) {
        float mv = __builtin_inff(); int mp = 0;
#pragma unroll
        for (int j = 0; j < KNN_K; ++j) {
            const int p = lane + 32 * j;
            const float v = sd[w][p];
            if (v < mv) { mv = v; mp = p; }
        }
        for (int off = 16; off; off >>= 1) {
            const float ov = __shfl_xor(mv, off, 32);
            const int   op = __shfl_xor(mp, off, 32);
            if (ov < mv || (ov == mv && op < mp)) { mv = ov; mp = op; }
        }
        if (lane == 0) {
            snbr[w][r] = si[w][mp];
            sd[w][mp] = __builtin_inff();
        }
        __syncthreads();
    }

    // gather-max over the K neighbors: lane owns 4 channels
    const int c0 = lane * 4;
    v4f mx;
    mx.x = -__builtin_inff(); mx.y = -__builtin_inff();
    mx.z = -__builtin_inff(); mx.w = -__builtin_inff();
#pragma unroll
    for (int k = 0; k < KNN_K; ++k) {
        const int nb = snbr[w][k];
        const v4f v = *(const v4f*)(h + nb * C_OUT + c0);
        mx.x = fmaxf(mx.x, v.x);
        mx.y = fmaxf(mx.y, v.y);
        mx.z = fmaxf(mx.z, v.z);
        mx.w = fmaxf(mx.w, v.w);
    }
    *(v4f*)(out + q * C_OUT + c0) = mx;
}

// ---------------------------------------------------------------------------
extern "C" void kernel_launch(void* const* d_in, const int* in_sizes, int n_in,
                              void* d_out, int out_size, void* d_ws, size_t ws_size,
                              hipStream_t stream)
{
    const float* x     = (const float*)d_in[0];   // [N, 64]
    const float* pos   = (const float*)d_in[1];   // [N, 3]
    // d_in[2] = batch (all zeros, unused beyond sub_batch output)
    const float* W     = (const float*)d_in[3];   // [64, 128]
    const float* bias  = (const float*)d_in[4];   // [128]
    const float* gamma = (const float*)d_in[5];   // [128]
    const float* beta  = (const float*)d_in[6];   // [128]

    float* out      = (float*)d_out;                    // [M, 128]
    float* sub_pos  = out + (size_t)M_PTS * C_OUT;      // [M, 3]
    int*   sub_batch = (int*)(sub_pos + (size_t)M_PTS * 3); // [M]

    float* h = (float*)d_ws;                            // [N, 128] scratch (16 MB)

    fps_kernel<<<1, 1024, 0, stream>>>(pos, sub_pos, sub_batch);
    mlp_ln_relu_kernel<<<N_PTS / 128, 256, 0, stream>>>(x, W, bias, gamma, beta, h);
    knn_pool_kernel<<<M_PTS / 8, 256, 0, stream>>>(pos, sub_pos, h, out);
}